// Top2Router_52441550684578
// MI455X (gfx1250) — compile-verified
//
#include <hip/hip_runtime.h>
#include <hip/hip_bf16.h>

// ---------------------------------------------------------------------------
// Top-2 noisy router for MI455X (gfx1250, wave32, WMMA).
//   logits = x @ W + b + noise;  (TAU = 1, NOISY_STD = 1)
//   outputs (flat, in order): top_idx[16384,2], top_w[16384,2],
//                             probs_full[16384,64], logits[16384,64]
// ---------------------------------------------------------------------------

#define TOKENS   16384
#define D_MODEL  2048
#define N_EXP    64
#define KCHUNK   128          // K rows of W staged in LDS per iteration
#define SSTR     (KCHUNK + 4) // transposed-LDS stride: even (b64 align) and
                              // 4n+k bank pattern -> conflict-free b64 reads
#define NOISY_STD 1.0f
#define TAU       1.0f

typedef __attribute__((ext_vector_type(2))) float v2f;
typedef __attribute__((ext_vector_type(8))) float v8f;

// ---------------------------------------------------------------------------
// Kernel 1: fp32 WMMA GEMM (16x16x4) + bias + noise -> logits
// Block = 256 threads = 8 waves. Wave w handles tokens [blk*128 + 16w, +16).
// W staged TRANSPOSED in LDS: sWT[n*SSTR + k] so the B-fragment pair
// (W[k,n], W[k+1,n]) is one aligned ds_load_b64 — no register shuffles.
// ---------------------------------------------------------------------------
__global__ __launch_bounds__(256) void
router_gemm_kernel(const float* __restrict__ x,
                   const float* __restrict__ noise,
                   const float* __restrict__ W,
                   const float* __restrict__ bias,
                   float* __restrict__ logits)
{
    __shared__ float sWT[N_EXP * SSTR];         // ~33 KB of 320 KB WGP LDS

    const int lane  = threadIdx.x & 31;
    const int wave  = threadIdx.x >> 5;
    const int laneM = lane & 15;                // row (A) / column (B) in 16
    const int hi    = lane >> 4;                // 0: K,K+1   1: K+2,K+3
    const int koff  = hi << 1;
    const int mbase = blockIdx.x * 128 + wave * 16;

    v8f acc[4] = {};                            // 16 tokens x 64 experts, f32

    const float* xrow = x + (size_t)(mbase + laneM) * D_MODEL + koff;

    // Per-tile LDS base for this lane's B column (constant over K).
    const float* bbase0 = &sWT[( 0 + laneM) * SSTR + koff];
    const float* bbase1 = &sWT[(16 + laneM) * SSTR + koff];
    const float* bbase2 = &sWT[(32 + laneM) * SSTR + koff];
    const float* bbase3 = &sWT[(48 + laneM) * SSTR + koff];

    for (int k0 = 0; k0 < D_MODEL; k0 += KCHUNK) {
        __syncthreads();                         // protect sWT reuse
        // Cooperative transpose-stage of W[k0 .. k0+KCHUNK) into LDS.
        const float4* wsrc = (const float4*)(W + (size_t)k0 * N_EXP);
        #pragma unroll 4
        for (int i = threadIdx.x; i < (KCHUNK * N_EXP) / 4; i += 256) {
            const float4 q = wsrc[i];
            const int k = i >> 4;                // (4i)/64
            const int n = (i & 15) << 2;         // (4i)%64
            sWT[(n + 0) * SSTR + k] = q.x;
            sWT[(n + 1) * SSTR + k] = q.y;
            sWT[(n + 2) * SSTR + k] = q.z;
            sWT[(n + 3) * SSTR + k] = q.w;
        }
        __syncthreads();

        #pragma unroll 4
        for (int kk = 0; kk < KCHUNK; kk += 4) {
            // A tile 16x4 fp32: lanes 0-15 hold (K,K+1), lanes 16-31 (K+2,K+3)
            const float2 av = *(const float2*)(xrow + k0 + kk);
            v2f a; a.x = av.x; a.y = av.y;

            // Four B tiles 4x16 fp32: one ds_load_b64 each (adjacent k pair).
            const float2 b0 = *(const float2*)(bbase0 + kk);
            const float2 b1 = *(const float2*)(bbase1 + kk);
            const float2 b2 = *(const float2*)(bbase2 + kk);
            const float2 b3 = *(const float2*)(bbase3 + kk);

            v2f vb;
            vb.x = b0.x; vb.y = b0.y;
            acc[0] = __builtin_amdgcn_wmma_f32_16x16x4_f32(
                false, a, false, vb, (short)0, acc[0], false, false);
            vb.x = b1.x; vb.y = b1.y;
            acc[1] = __builtin_amdgcn_wmma_f32_16x16x4_f32(
                false, a, false, vb, (short)0, acc[1], false, false);
            vb.x = b2.x; vb.y = b2.y;
            acc[2] = __builtin_amdgcn_wmma_f32_16x16x4_f32(
                false, a, false, vb, (short)0, acc[2], false, false);
            vb.x = b3.x; vb.y = b3.y;
            acc[3] = __builtin_amdgcn_wmma_f32_16x16x4_f32(
                false, a, false, vb, (short)0, acc[3], false, false);
        }
    }

    // Epilogue: C/D layout -> VGPR r holds row M = r + 8*hi, N = 16t + laneM.
    const float inv_tau = 1.0f / TAU;
    #pragma unroll
    for (int t = 0; t < 4; ++t) {
        const int e = t * 16 + laneM;
        const float be = bias[e];
        #pragma unroll
        for (int r = 0; r < 8; ++r) {
            const int tok = mbase + r + 8 * hi;
            float v = acc[t][r] + be + noise[(size_t)tok * N_EXP + e] * NOISY_STD;
            logits[(size_t)tok * N_EXP + e] = v * inv_tau;
        }
    }
}

// ---------------------------------------------------------------------------
// Kernel 2: per-token top-2 + softmax(2) + softmax(64). One thread per token.
// ---------------------------------------------------------------------------
__global__ __launch_bounds__(256) void
router_topk_kernel(const float* __restrict__ logits,
                   float* __restrict__ top_idx,   // written as float
                   float* __restrict__ top_w,
                   float* __restrict__ probs)
{
    const int tok = blockIdx.x * 256 + threadIdx.x;
    if (tok >= TOKENS) return;

    const float* lg = logits + (size_t)tok * N_EXP;
    float v[N_EXP];
    #pragma unroll
    for (int i = 0; i < N_EXP / 4; ++i) {
        float4 q = ((const float4*)lg)[i];
        v[4 * i + 0] = q.x; v[4 * i + 1] = q.y;
        v[4 * i + 2] = q.z; v[4 * i + 3] = q.w;
    }

    // top-1 (strict > => lowest index on ties, matching jax.lax.top_k)
    int   i0 = 0; float v0 = v[0];
    #pragma unroll
    for (int i = 1; i < N_EXP; ++i)
        if (v[i] > v0) { v0 = v[i]; i0 = i; }

    // top-2
    int   i1 = (i0 == 0) ? 1 : 0; float v1 = v[i1];
    #pragma unroll
    for (int i = 0; i < N_EXP; ++i)
        if (i != i0 && v[i] > v1) { v1 = v[i]; i1 = i; }

    // softmax over the two selected (v0 >= v1)
    const float e1    = __expf(v1 - v0);
    const float denom = 1.0f + e1;
    top_idx[2 * tok + 0] = (float)i0;
    top_idx[2 * tok + 1] = (float)i1;
    top_w[2 * tok + 0]   = 1.0f / denom;
    top_w[2 * tok + 1]   = e1 / denom;

    // full softmax over all 64 experts (max = v0)
    float s = 0.0f;
    #pragma unroll
    for (int i = 0; i < N_EXP; ++i) { v[i] = __expf(v[i] - v0); s += v[i]; }
    const float inv = 1.0f / s;
    float* pr = probs + (size_t)tok * N_EXP;
    #pragma unroll
    for (int i = 0; i < N_EXP / 4; ++i) {
        float4 q;
        q.x = v[4 * i + 0] * inv; q.y = v[4 * i + 1] * inv;
        q.z = v[4 * i + 2] * inv; q.w = v[4 * i + 3] * inv;
        ((float4*)pr)[i] = q;
    }
}

// ---------------------------------------------------------------------------
extern "C" void kernel_launch(void* const* d_in, const int* in_sizes, int n_in,
                              void* d_out, int out_size, void* d_ws, size_t ws_size,
                              hipStream_t stream)
{
    const float* x     = (const float*)d_in[0];   // [16384, 2048]
    const float* noise = (const float*)d_in[1];   // [16384, 64]
    const float* W     = (const float*)d_in[2];   // [2048, 64]
    const float* bias  = (const float*)d_in[3];   // [64]

    float* out      = (float*)d_out;
    float* top_idx  = out;                                   // 16384*2
    float* top_w    = out + (size_t)TOKENS * 2;              // 16384*2
    float* probs    = out + (size_t)TOKENS * 4;              // 16384*64
    float* logits   = out + (size_t)TOKENS * (4 + N_EXP);    // 16384*64

    router_gemm_kernel<<<TOKENS / 128, 256, 0, stream>>>(x, noise, W, bias, logits);
    router_topk_kernel<<<TOKENS / 256, 256, 0, stream>>>(logits, top_idx, top_w, probs);
}